// BCQWeightQuantizer_17068200034871
// MI455X (gfx1250) — compile-verified
//
#include <hip/hip_runtime.h>

typedef float v2f __attribute__((ext_vector_type(2)));
typedef float v4f __attribute__((ext_vector_type(4)));
typedef float v8f __attribute__((ext_vector_type(8)));

#define WAVES_PER_BLOCK 8
#define GROUPS_PER_WAVE 16
#define GROUPS_PER_BLOCK (WAVES_PER_BLOCK * GROUPS_PER_WAVE)  // 128
#define GSIZE 128   // elements per group
#define NLEV 16     // 2^4 BCQ levels

// One wave handles 16 consecutive groups:
//   1) one V_WMMA_F32_16X16X4_F32 builds the 16x16 level table alpha @ grid.T
//   2) level table parked in LDS (per-wave 1 KB slice)
//   3) stream 16x128 elements, snapping each to the nearest level.
__global__ __launch_bounds__(256) void bcq_fwd_kernel(
    const float* __restrict__ w,
    const float* __restrict__ scale,
    const float* __restrict__ zp,
    const float* __restrict__ alpha,
    const float* __restrict__ shift,
    float* __restrict__ out,
    int nGroups)
{
    __shared__ __align__(16) float lev[WAVES_PER_BLOCK * 256];

    const int lane = threadIdx.x & 31;
    const int wave = __builtin_amdgcn_readfirstlane((int)(threadIdx.x >> 5));
    const int gw   = (blockIdx.x * WAVES_PER_BLOCK + wave) * GROUPS_PER_WAVE;
    if (gw >= nGroups) return;

    // ---- Level table via WMMA: D[M=group][N=level] = sum_k A[M][k] * B[k][N]
    // A layout (32-bit A 16x4): lanes 0-15 -> M=row, VGPR0/1 = K0/K1;
    //                           lanes 16-31 -> M=row, VGPR0/1 = K2/K3.
    const int row  = lane & 15;
    const int half = lane >> 4;            // 0: K={0,1}, 1: K={2,3}

    const v2f* a2 = reinterpret_cast<const v2f*>(alpha);   // [G][2] of v2f
    v2f A = a2[(size_t)(gw + row) * 2 + half];

    // B[k][n] = grid[n][k] = 1 - 2*((n >> (3-k)) & 1); mirrored lane-half split.
    const int k0 = half * 2;
    v2f B;
    B[0] = 1.0f - 2.0f * (float)((row >> (3 - k0)) & 1);
    B[1] = 1.0f - 2.0f * (float)((row >> (2 - k0)) & 1);

    v8f C = {};
    C = __builtin_amdgcn_wmma_f32_16x16x4_f32(
        /*neg_a=*/false, A, /*neg_b=*/false, B,
        /*c_mod=*/(short)0, C, /*reuse_a=*/false, /*reuse_b=*/false);

    // D layout: VGPR r, lanes 0-15 -> M=r, lanes 16-31 -> M=r+8; N = lane&15.
    float* myLev = &lev[wave * 256];
#pragma unroll
    for (int r = 0; r < 8; ++r) {
        myLev[(r + half * 8) * NLEV + row] = C[r];
    }
    __syncthreads();

    const float* wBase = w   + (size_t)gw * GSIZE;
    float*       oBase = out + (size_t)gw * GSIZE;

    for (int i = 0; i < GROUPS_PER_WAVE; ++i) {
        const int g = gw + i;                       // wave-uniform -> s_load scalars
        const float sc = scale[g];
        const float z  = zp[g];
        const float sh = shift[g];
        const float rs = 1.0f / sc;                 // hoisted divide
        const float c0 = z - sh - 7.5f;             // xi  = fma(w, rs, c0)
        const float c1 = (sh + 7.5f - z) * sc;      // out = fma(q, sc, c1)

        float L[NLEV];
#pragma unroll
        for (int l4 = 0; l4 < 4; ++l4) {            // 4x ds_load_b128, broadcast
            v4f t = ((const v4f*)(myLev + i * NLEV))[l4];
            L[l4 * 4 + 0] = t.x; L[l4 * 4 + 1] = t.y;
            L[l4 * 4 + 2] = t.z; L[l4 * 4 + 3] = t.w;
        }

        // 32 lanes x v4f = whole group (128 elements), coalesced + streaming.
        v4f wv = __builtin_nontemporal_load(
            ((const v4f*)(wBase + (size_t)i * GSIZE)) + lane);
        float xs[4] = {wv.x, wv.y, wv.z, wv.w};
        float o[4];
#pragma unroll
        for (int e = 0; e < 4; ++e) {
            float xi   = fmaf(xs[e], rs, c0);
            float best = xi - L[0];
            float q    = L[0];
#pragma unroll
            for (int l = 1; l < NLEV; ++l) {
                float d  = xi - L[l];
                // |d| < |best|: abs folds into v_cmp source modifiers (free);
                // strict < keeps the first minimum, matching jnp.argmin.
                bool cnd = __builtin_fabsf(d) < __builtin_fabsf(best);
                best = cnd ? d    : best;
                q    = cnd ? L[l] : q;
            }
            o[e] = fmaf(q, sc, c1);
        }
        v4f ov; ov.x = o[0]; ov.y = o[1]; ov.z = o[2]; ov.w = o[3];
        __builtin_nontemporal_store(ov, ((v4f*)(oBase + (size_t)i * GSIZE)) + lane);
    }
}

extern "C" void kernel_launch(void* const* d_in, const int* in_sizes, int n_in,
                              void* d_out, int out_size, void* d_ws, size_t ws_size,
                              hipStream_t stream) {
    const float* w     = (const float*)d_in[0];
    const float* scale = (const float*)d_in[1];
    const float* zp    = (const float*)d_in[2];
    const float* alpha = (const float*)d_in[3];
    const float* shift = (const float*)d_in[4];
    float* out = (float*)d_out;

    const int G = in_sizes[1];  // scale has one entry per group
    const int blocks = (G + GROUPS_PER_BLOCK - 1) / GROUPS_PER_BLOCK;
    bcq_fwd_kernel<<<blocks, 256, 0, stream>>>(w, scale, zp, alpha, shift, out, G);
}